// UltraMem_18708877541525
// MI455X (gfx1250) — compile-verified
//
#include <hip/hip_runtime.h>
#include <hip/hip_bf16.h>
#include <float.h>

// ---- problem constants ----
#define BB      4
#define NN      2048
#define DIMC    512
#define DQKC    128
#define HEADSC  2
#define KK      32
#define NKEYS   1000
#define VEC     4
#define NVIRT   250000
#define DVC     256
#define KCONVC  5
#define BNROWS  (BB*NN)            // 8192
#define RMS_EPS 1.1920929e-7f
#define LN_EPS  1e-5f

typedef __attribute__((ext_vector_type(2))) float v2f;
typedef __attribute__((ext_vector_type(8))) float v8f;

static __device__ __forceinline__ v8f wmma4(v2f a, v2f b, v8f c) {
  // V_WMMA_F32_16X16X4_F32 : D = A(16x4) * B(4x16) + C(16x16), fp32 native
  return __builtin_amdgcn_wmma_f32_16x16x4_f32(false, a, false, b, (short)0, c,
                                               false, false);
}

// LDS byte offset of a pointer into __shared__ (for async-to-LDS VDST operand)
static __device__ __forceinline__ unsigned lds_off(const void* p) {
  return (unsigned)(unsigned long long)(const __attribute__((address_space(3))) void*)p;
}

// async copy: 16 bytes global -> LDS (per lane), tracked by ASYNCcnt
static __device__ __forceinline__ void async_b128(unsigned lds, const void* g) {
  asm volatile("global_load_async_to_lds_b128 %0, %1, off"
               :: "v"(lds), "v"((unsigned long long)g) : "memory");
}
// async copy: 4 bytes global -> LDS (per lane)
static __device__ __forceinline__ void async_b32(unsigned lds, const void* g) {
  asm volatile("global_load_async_to_lds_b32 %0, %1, off"
               :: "v"(lds), "v"((unsigned long long)g) : "memory");
}
static __device__ __forceinline__ void wait_async0() {
  asm volatile("s_wait_asynccnt 0x0" ::: "memory");
}
static __device__ __forceinline__ void wait_async1() {
  asm volatile("s_wait_asynccnt 0x1" ::: "memory");
}

// ---------------- tiny 2x2 SVD of core (per head) ----------------
__global__ void k_svd(const float* __restrict__ core, float* __restrict__ uv) {
  if (threadIdx.x != 0 || blockIdx.x != 0) return;
  for (int h = 0; h < HEADSC; ++h) {
    float a = core[h*4+0], b = core[h*4+1], c = core[h*4+2], d = core[h*4+3];
    float p = a*a + c*c, q = a*b + c*d, s = b*b + d*d;          // A^T A
    float half = (p - s) * 0.5f;
    float lam = (p + s) * 0.5f + sqrtf(fmaxf(0.f, half*half + q*q));
    float e0 = lam - s, e1 = q;
    float f0 = q, f1 = lam - p;
    float v0, v1;
    if (e0*e0 + e1*e1 >= f0*f0 + f1*f1) { v0 = e0; v1 = e1; }
    else                                 { v0 = f0; v1 = f1; }
    float nv = sqrtf(v0*v0 + v1*v1);
    if (nv < 1e-30f) { v0 = 1.f; v1 = 0.f; } else { v0 /= nv; v1 /= nv; }
    float u0 = a*v0 + b*v1, u1 = c*v0 + d*v1;                    // u = A v / |Av|
    float nu = sqrtf(u0*u0 + u1*u1);
    if (nu < 1e-30f) { u0 = 1.f; u1 = 0.f; } else { u0 /= nu; u1 /= nu; }
    float mu = (fabsf(u0) >= fabsf(u1)) ? u0 : u1;               // sign convention
    if (mu < 0.f) { u0 = -u0; u1 = -u1; v0 = -v0; v1 = -v1; }
    uv[0 + h*2 + 0] = u0; uv[0 + h*2 + 1] = u1;                  // u_vec
    uv[4 + h*2 + 0] = v0; uv[4 + h*2 + 1] = v1;                  // t_vec
  }
}

// ---------------- RMSNorm over DIM=512 ----------------
__global__ void k_rms(const float* __restrict__ tok, const float* __restrict__ w,
                      float* __restrict__ xn) {
  __shared__ float red[256];
  const int row = blockIdx.x, t = threadIdx.x;
  const float* in = tok + (size_t)row * DIMC;
  float v0 = in[t], v1 = in[t + 256];
  red[t] = v0*v0 + v1*v1;
  __syncthreads();
  for (int s = 128; s > 0; s >>= 1) { if (t < s) red[t] += red[t+s]; __syncthreads(); }
  float scale = rsqrtf(red[0] / (float)DIMC + RMS_EPS);
  float* out = xn + (size_t)row * DIMC;
  out[t]       = v0 * scale * w[t];
  out[t + 256] = v1 * scale * w[t + 256];
}

// ---------------- depthwise causal conv (KCONV=5) ----------------
__global__ void k_conv(const float* __restrict__ xn, const float* __restrict__ cw,
                       const float* __restrict__ cb, float* __restrict__ qin) {
  int idx = blockIdx.x * blockDim.x + threadIdx.x;      // over B*N*DIM
  int c = idx & (DIMC - 1);
  int nb = idx >> 9;
  int n = nb & (NN - 1);
  int b = nb >> 11;
  float acc = cb[c];
  #pragma unroll
  for (int tp = 0; tp < KCONVC; ++tp) {
    int np = n - (KCONVC - 1) + tp;
    if (np >= 0)
      acc += xn[((size_t)b * NN + np) * DIMC + c] * cw[c * KCONVC + tp];
  }
  qin[idx] = acc;
}

// ---- GEMM: qpre = qin(8192x512) @ wq(512x128); 8 waves share async LDS B-panel
__global__ void k_gemm_q(const float* __restrict__ A, const float* __restrict__ Bw,
                         float* __restrict__ C) {
  __shared__ float Bs[DIMC * 16];               // 512x16 f32 panel = 32 KB
  const int t = threadIdx.x;
  const int wave = t >> 5, lane = t & 31;
  const int ml = lane & 15, hi = lane >> 4;
  const int nt = blockIdx.y;
  #pragma unroll
  for (int it = 0; it < 8; ++it) {              // 2048 x 16B chunks
    int qd = it * 256 + t;
    int k = qd >> 2, part = qd & 3;
    async_b128(lds_off(&Bs[qd * 4]), Bw + (size_t)k * DQKC + nt * 16 + part * 4);
  }
  wait_async0();
  __syncthreads();
  const int m = (blockIdx.x * 8 + wave) * 16 + ml;
  v8f acc = {};
  for (int k0 = 0; k0 < DIMC; k0 += 4) {
    int ka = k0 + 2 * hi;
    v2f a, b;
    a.x = A[(size_t)m * DIMC + ka];  a.y = A[(size_t)m * DIMC + ka + 1];
    b.x = Bs[ka * 16 + ml];          b.y = Bs[(ka + 1) * 16 + ml];
    acc = wmma4(a, b, acc);
  }
  #pragma unroll
  for (int i = 0; i < 8; ++i)
    C[(size_t)((blockIdx.x * 8 + wave) * 16 + i + 8 * hi) * DQKC + nt * 16 + ml] = acc[i];
}

// ---------------- LayerNorm (no bias) over 128 cols ----------------
__global__ void k_ln128(const float* __restrict__ in, const float* __restrict__ w,
                        float* __restrict__ out) {
  __shared__ float red[128];
  const int row = blockIdx.x, t = threadIdx.x;
  float x = in[(size_t)row * DQKC + t];
  red[t] = x; __syncthreads();
  for (int s = 64; s > 0; s >>= 1) { if (t < s) red[t] += red[t+s]; __syncthreads(); }
  float mu = red[0] / (float)DQKC;
  __syncthreads();
  float dx = x - mu;
  red[t] = dx * dx; __syncthreads();
  for (int s = 64; s > 0; s >>= 1) { if (t < s) red[t] += red[t+s]; __syncthreads(); }
  float var = red[0] / (float)DQKC;
  out[(size_t)row * DQKC + t] = dx * rsqrtf(var + LN_EPS) * w[t];
}

// ------- fold u/t vectors into normalized keys: kuv[s][h][m][d] -------
__global__ void k_fold(const float* __restrict__ keysn, const float* __restrict__ uv,
                       float* __restrict__ kuv) {
  int idx = blockIdx.x * blockDim.x + threadIdx.x;  // 2*2*1000*128
  int d = idx & 127;
  int m = (idx >> 7) % NKEYS;
  int hs = idx / (128 * NKEYS);
  int h = hs & 1, s = hs >> 1;
  float w0 = uv[s * 4 + h * 2 + 0], w1 = uv[s * 4 + h * 2 + 1];
  float k0 = keysn[((size_t)(s * 2 + 0) * NKEYS + m) * DQKC + d];
  float k1 = keysn[((size_t)(s * 2 + 1) * NKEYS + m) * DQKC + d];
  kuv[((size_t)(s * 2 + h) * NKEYS + m) * DQKC + d] = w0 * k0 + w1 * k1;
}

// ------- GEMM: mscore[z] = queries(8192x128) @ kuv[z](1000x128)^T -------
// 8-wave workgroup; 16-key B panel async-staged to LDS (stride 132 kills conflicts)
#define BSTR 132
__global__ void k_gemm_m(const float* __restrict__ Q, const float* __restrict__ KUV,
                         float* __restrict__ MS) {
  __shared__ float Bs[16 * BSTR];               // ~8.25 KB
  const int t = threadIdx.x;
  const int wave = t >> 5, lane = t & 31;
  const int ml = lane & 15, hi = lane >> 4;
  const int nt = blockIdx.y, z = blockIdx.z;
  const int n0 = nt * 16;
  const float* Bk = KUV + (size_t)z * NKEYS * DQKC;
  #pragma unroll
  for (int it = 0; it < 2; ++it) {              // 512 x 16B chunks (16 keys x 128 f32)
    int qd = it * 256 + t;
    int kk = qd >> 5, part = qd & 31;
    int key = n0 + kk; if (key >= NKEYS) key = NKEYS - 1;   // clamp (cols masked at store)
    async_b128(lds_off(&Bs[kk * BSTR + part * 4]),
               Bk + (size_t)key * DQKC + part * 4);
  }
  wait_async0();
  __syncthreads();
  const int m = (blockIdx.x * 8 + wave) * 16 + ml;
  v8f acc = {};
  for (int k0 = 0; k0 < DQKC; k0 += 4) {
    int ka = k0 + 2 * hi;
    v2f a, b;
    a.x = Q[(size_t)m * DQKC + ka];  a.y = Q[(size_t)m * DQKC + ka + 1];
    b.x = Bs[ml * BSTR + ka];        b.y = Bs[ml * BSTR + ka + 1];
    acc = wmma4(a, b, acc);
  }
  int col = n0 + ml;
  if (col < NKEYS) {
    #pragma unroll
    for (int i = 0; i < 8; ++i)
      MS[((size_t)z * BNROWS + (blockIdx.x * 8 + wave) * 16 + i + 8 * hi) * NKEYS + col]
        = acc[i];
  }
}

// ---------------- top-32 of 1000 (jax tie-break: lower index) ----------------
__global__ void k_topk(const float* __restrict__ MS, int* __restrict__ TOPI) {
  __shared__ float vals[NKEYS];
  __shared__ float rv[128]; __shared__ int ri[128];
  const int rowb = blockIdx.x, t = threadIdx.x;
  const float* src = MS + (size_t)rowb * NKEYS;
  for (int p = t; p < NKEYS; p += 128) vals[p] = src[p];
  __syncthreads();
  for (int it = 0; it < KK; ++it) {
    float bv = -FLT_MAX; int bi = 0x7fffffff;
    for (int p = t; p < NKEYS; p += 128)
      if (vals[p] > bv) { bv = vals[p]; bi = p; }
    rv[t] = bv; ri[t] = bi;
    __syncthreads();
    for (int s = 64; s > 0; s >>= 1) {
      if (t < s) {
        if (rv[t+s] > rv[t] || (rv[t+s] == rv[t] && ri[t+s] < ri[t])) {
          rv[t] = rv[t+s]; ri[t] = ri[t+s];
        }
      }
      __syncthreads();
    }
    if (t == 0) { TOPI[(size_t)rowb * KK + it] = ri[0]; vals[ri[0]] = -FLT_MAX; }
    __syncthreads();
  }
}

// ---- scores = fr*core*fc^T, top-32 of 1024, relu, index math ----
__global__ void k_scores(const float* __restrict__ QRY, const float* __restrict__ KEYSN,
                         const float* __restrict__ core, const int* __restrict__ rpm,
                         const int* __restrict__ TOPI, float* __restrict__ FSC,
                         int* __restrict__ EXPI, int* __restrict__ memOut) {
  __shared__ float q[DQKC];
  __shared__ int tr[KK], tc[KK];
  __shared__ float fr[2*KK], fc[2*KK];
  __shared__ float sc[KK*KK];
  __shared__ float rv[256]; __shared__ int ri[256];
  const int hb = blockIdx.x, t = threadIdx.x;
  const int h = hb >> 13, bn = hb & (BNROWS - 1);
  if (t < DQKC) q[t] = QRY[(size_t)bn * DQKC + t];
  if (t < KK)            tr[t]      = TOPI[((size_t)(0*2 + h) * BNROWS + bn) * KK + t];
  else if (t < 2*KK)     tc[t - KK] = TOPI[((size_t)(1*2 + h) * BNROWS + bn) * KK + (t - KK)];
  __syncthreads();
  if (t < 128) {
    int which = t >> 6;            // 0 -> fr (row keys), 1 -> fc (col keys)
    int sub = t & 63;
    int i = sub >> 1, r = sub & 1;
    int m = which == 0 ? tr[i] : tc[i];
    const float* kp = KEYSN + ((size_t)(which * 2 + r) * NKEYS + m) * DQKC;
    float dsum = 0.f;
    for (int dd = 0; dd < DQKC; ++dd) dsum += q[dd] * kp[dd];
    if (which == 0) fr[i*2 + r] = dsum; else fc[i*2 + r] = dsum;
  }
  __syncthreads();
  float c00 = core[h*4+0], c01 = core[h*4+1], c10 = core[h*4+2], c11 = core[h*4+3];
  #pragma unroll
  for (int u = 0; u < 4; ++u) {
    int p = t * 4 + u;
    int i = p >> 5, j = p & 31;
    sc[p] = fr[i*2]   * (c00 * fc[j*2] + c01 * fc[j*2+1])
          + fr[i*2+1] * (c10 * fc[j*2] + c11 * fc[j*2+1]);
  }
  __syncthreads();
  for (int it = 0; it < KK; ++it) {
    float bv = -FLT_MAX; int bi = 0x7fffffff;
    #pragma unroll
    for (int u = 0; u < 4; ++u) {
      int p = t * 4 + u;
      if (sc[p] > bv) { bv = sc[p]; bi = p; }
    }
    rv[t] = bv; ri[t] = bi;
    __syncthreads();
    for (int s = 128; s > 0; s >>= 1) {
      if (t < s) {
        if (rv[t+s] > rv[t] || (rv[t+s] == rv[t] && ri[t+s] < ri[t])) {
          rv[t] = rv[t+s]; ri[t] = ri[t+s];
        }
      }
      __syncthreads();
    }
    if (t == 0) {
      int p = ri[0];
      int fi = tr[p >> 5] * NKEYS + tc[p & 31];
      FSC[(size_t)hb * KK + it] = fmaxf(rv[0], 0.f);
      memOut[(size_t)hb * KK + it] = fi >> 2;        // // VE
      EXPI[(size_t)hb * KK + it] = rpm[fi];
      sc[p] = -FLT_MAX;
    }
    __syncthreads();
  }
}

// ---- gather memories, scale, route to expert: async double-buffered in LDS ----
__global__ void k_pool(const float* __restrict__ MEM, const float* __restrict__ FSC,
                       const int* __restrict__ EXPI, const int* __restrict__ memIdx,
                       float* __restrict__ POOL) {
  __shared__ float pl[VEC * DVC];
  __shared__ float rowbuf[2][DVC];
  const int hb = blockIdx.x, t = threadIdx.x;
  const int h = hb >> 13;
  pl[t] = 0.f; pl[t + 256] = 0.f; pl[t + 512] = 0.f; pl[t + 768] = 0.f;
  const size_t base = (size_t)hb * KK;
  const float* memH = MEM + (size_t)h * NVIRT * DVC;
  // prologue: async-stage row 0 (thread t owns element t, written by its own wave)
  {
    int mi0 = memIdx[base];
    async_b32(lds_off(&rowbuf[0][t]), memH + (size_t)mi0 * DVC + t);
  }
  int buf = 0;
  for (int k = 0; k < KK; ++k) {
    if (k + 1 < KK) {
      int mi1 = memIdx[base + k + 1];
      async_b32(lds_off(&rowbuf[buf ^ 1][t]), memH + (size_t)mi1 * DVC + t);
      wait_async1();                   // row k landed (async loads complete in order)
    } else {
      wait_async0();
    }
    float s = FSC[base + k];
    int e = EXPI[base + k];
    pl[e * DVC + t] += rowbuf[buf][t] * s;   // thread t owns lanes t mod 256
    buf ^= 1;
  }
  float* dst = POOL + (size_t)hb * (VEC * DVC);
  dst[t] = pl[t]; dst[t+256] = pl[t+256]; dst[t+512] = pl[t+512]; dst[t+768] = pl[t+768];
}

// ---- final GEMM: out = pooled(16384x1024) @ vproj_flat(1024x256) ----
// 8-wave workgroup; full 1024x16 B-panel (64 KB) async-staged once into LDS
__global__ void k_gemm_agg(const float* __restrict__ POOL, const float* __restrict__ VP,
                           float* __restrict__ outF) {
  __shared__ float Bs[(VEC*DVC) * 16];          // 1024x16 f32 = 64 KB
  const int t = threadIdx.x;
  const int wave = t >> 5, lane = t & 31;
  const int ml = lane & 15, hi = lane >> 4;
  const int nt = blockIdx.y;
  #pragma unroll
  for (int it = 0; it < 16; ++it) {             // 4096 x 16B chunks
    int qd = it * 256 + t;
    int k = qd >> 2, part = qd & 3;
    async_b128(lds_off(&Bs[qd * 4]), VP + (size_t)k * DVC + nt * 16 + part * 4);
  }
  wait_async0();
  __syncthreads();
  const int m = (blockIdx.x * 8 + wave) * 16 + ml;
  v8f acc = {};
  for (int k0 = 0; k0 < VEC * DVC; k0 += 4) {
    int ka = k0 + 2 * hi;
    v2f a, b;
    a.x = POOL[(size_t)m * (VEC*DVC) + ka];  a.y = POOL[(size_t)m * (VEC*DVC) + ka + 1];
    b.x = Bs[ka * 16 + ml];                  b.y = Bs[(ka + 1) * 16 + ml];
    acc = wmma4(a, b, acc);
  }
  #pragma unroll
  for (int i = 0; i < 8; ++i) {
    int row = (blockIdx.x * 8 + wave) * 16 + i + 8 * hi;   // h*8192 + bn
    int h = row >> 13, bn = row & (BNROWS - 1);
    outF[(size_t)bn * (HEADSC*DVC) + h * DVC + nt * 16 + ml] = acc[i];
  }
}

// ---------------- host side ----------------
extern "C" void kernel_launch(void* const* d_in, const int* in_sizes, int n_in,
                              void* d_out, int out_size, void* d_ws, size_t ws_size,
                              hipStream_t stream) {
  (void)in_sizes; (void)n_in; (void)out_size; (void)ws_size;
  const float* tokens = (const float*)d_in[0];
  const float* rms_w  = (const float*)d_in[1];
  const float* conv_w = (const float*)d_in[2];
  const float* conv_b = (const float*)d_in[3];
  const float* wq     = (const float*)d_in[4];
  const float* qln_w  = (const float*)d_in[5];
  const float* kln_w  = (const float*)d_in[6];
  const float* keys_p = (const float*)d_in[7];
  const float* core   = (const float*)d_in[8];
  const float* vproj  = (const float*)d_in[9];
  const float* mems   = (const float*)d_in[10];
  const int*   rpm    = (const int*)d_in[11];

  float* W = (float*)d_ws;
  // workspace layout (float offsets); pooled aliases dead mscore
  const size_t OFF_XN   = 0;
  const size_t OFF_QIN  = OFF_XN   + (size_t)BNROWS * DIMC;
  const size_t OFF_QPRE = OFF_QIN  + (size_t)BNROWS * DIMC;
  const size_t OFF_QRY  = OFF_QPRE + (size_t)BNROWS * DQKC;
  const size_t OFF_KN   = OFF_QRY  + (size_t)BNROWS * DQKC;
  const size_t OFF_KUV  = OFF_KN   + (size_t)4 * NKEYS * DQKC;
  const size_t OFF_UV   = OFF_KUV  + (size_t)4 * NKEYS * DQKC;
  const size_t OFF_MS   = OFF_UV   + 64;
  const size_t OFF_POOL = OFF_MS;                    // alias (mscore dead by then)
  const size_t OFF_TOPI = OFF_MS   + (size_t)4 * BNROWS * NKEYS;
  const size_t OFF_FSC  = OFF_TOPI + (size_t)4 * BNROWS * KK;
  const size_t OFF_EXP  = OFF_FSC  + (size_t)HEADSC * BNROWS * KK;

  float* xn    = W + OFF_XN;
  float* qin   = W + OFF_QIN;
  float* qpre  = W + OFF_QPRE;
  float* qry   = W + OFF_QRY;
  float* keysn = W + OFF_KN;
  float* kuv   = W + OFF_KUV;
  float* uv    = W + OFF_UV;
  float* ms    = W + OFF_MS;
  float* pool  = W + OFF_POOL;
  int*   topI  = (int*)(W + OFF_TOPI);
  float* fsc   = W + OFF_FSC;
  int*   expI  = (int*)(W + OFF_EXP);

  float* outF = (float*)d_out;                                   // (B,N,H*DV) f32
  int*   memOut = (int*)d_out + (size_t)BB * NN * HEADSC * DVC;  // (H,B,N,K) i32

  k_svd  <<<1, 1, 0, stream>>>(core, uv);
  k_rms  <<<BNROWS, 256, 0, stream>>>(tokens, rms_w, xn);
  k_conv <<<(BNROWS * DIMC) / 256, 256, 0, stream>>>(xn, conv_w, conv_b, qin);
  k_gemm_q<<<dim3(BNROWS/128, DQKC/16), 256, 0, stream>>>(qin, wq, qpre);
  k_ln128<<<BNROWS, 128, 0, stream>>>(qpre, qln_w, qry);
  k_ln128<<<4 * NKEYS, 128, 0, stream>>>(keys_p, kln_w, keysn);
  k_fold <<<(4 * NKEYS * DQKC) / 256, 256, 0, stream>>>(keysn, uv, kuv);
  k_gemm_m<<<dim3(BNROWS/128, (NKEYS + 15)/16, 4), 256, 0, stream>>>(qry, kuv, ms);
  k_topk <<<4 * BNROWS, 128, 0, stream>>>(ms, topI);
  k_scores<<<HEADSC * BNROWS, 256, 0, stream>>>(qry, keysn, core, rpm, topI,
                                                fsc, expI, memOut);
  k_pool <<<HEADSC * BNROWS, 256, 0, stream>>>(mems, fsc, expI, memOut, pool);
  k_gemm_agg<<<dim3(HEADSC*BNROWS/128, DVC/16), 256, 0, stream>>>(pool, vproj, outF);
}